// ResGCN10_58128087384886
// MI455X (gfx1250) — compile-verified
//
#include <hip/hip_runtime.h>
#include <math.h>

typedef float v2f __attribute__((ext_vector_type(2)));
typedef float v8f __attribute__((ext_vector_type(8)));

#define NN      50000
#define NE      1600000
#define NFEAT   500
#define NHID    128
#define NCLASS  40
#define NCAT    (9 * NHID)              // 1152
#define MTILES  ((NN + 15) / 16)        // 3125 row tiles (NN % 16 == 0)
#define NPAD40  48                      // classifier columns padded to 3 tiles

// ---------------------------------------------------------------------------
// B repack: Bp[(q*NP + col)*4 + r] = (col < N) ? B[(4q + r)*N + col] : 0
// Turns each lane's WMMA B-fragment (rows k0+kk, k0+kk+1 at one column) into
// two adjacent floats -> one branch-free global_load_b64 in the GEMM loop.
// ---------------------------------------------------------------------------
__global__ void repack_B_kernel(const float* __restrict__ B, int K, int N, int NP,
                                float* __restrict__ Bp)
{
    const long long total = (long long)(K / 4) * NP * 4;
    long long i = (long long)blockIdx.x * blockDim.x + threadIdx.x;
    if (i >= total) return;
    const int       r   = (int)(i & 3);
    const long long qc  = i >> 2;
    const int       col = (int)(qc % NP);
    const int       q   = (int)(qc / NP);
    Bp[i] = (col < N) ? B[(long long)(4 * q + r) * N + col] : 0.0f;
}

// ---------------------------------------------------------------------------
// fp32 WMMA GEMM:  C[M, NSTORE] = A[M, KC] @ B (+ bias optional)
// One wave computes a 16 x (NT*16) strip with V_WMMA_F32_16X16X4_F32.
// NT, KC, NSTORE, NPc are compile-time: branch-free inner loop, immediate-
// offset addressing, and software-pipelined (double-buffered) fragments so
// loads for quad q+1 overlap the WMMAs of quad q.
// ---------------------------------------------------------------------------
template <int NT, int KC, int NSTORE, int NPc>
__global__ __launch_bounds__(256) void gemm_wmma_f32(
    const float* __restrict__ A, int lda,
    const float* __restrict__ Bp,
    float* __restrict__ C, int ldc,
    const float* __restrict__ bias)
{
    const int lane  = threadIdx.x & 31;
    const int wave  = threadIdx.x >> 5;
    const int rtile = blockIdx.x * 8 + wave;
    if (rtile >= MTILES) return;                 // uniform per wave: EXEC stays full

    constexpr int KQ = KC / 4;                   // K quads
    const int m0 = rtile * 16;
    const int n0 = blockIdx.y * (NT * 16);
    const int mr = lane & 15;                    // A row-in-tile / B,C column-in-tile
    const int kk = (lane >> 4) * 2;              // lanes 0-15 -> K 0,1 ; 16-31 -> K 2,3

    v8f acc[NT];
#pragma unroll
    for (int t = 0; t < NT; ++t)
#pragma unroll
        for (int v = 0; v < 8; ++v) acc[t][v] = 0.0f;

    const float* Arow = A + (long long)(m0 + mr) * lda + kk;    // +q*4 floats per quad
    const float* Bb   = Bp + (long long)(n0 + mr) * 4 + kk;     // +q*NPc*4, +t*64 floats

    // ---- prologue: fragments for quad 0
    v2f a0 = *(const v2f*)(Arow);
    v2f b0[NT];
#pragma unroll
    for (int t = 0; t < NT; ++t)
        b0[t] = *(const v2f*)(Bb + t * 64);

    // ---- pipelined main loop: issue loads for q+1, then WMMA quad q
#pragma unroll 2
    for (int q = 0; q < KQ; ++q) {
        const int qn = (q + 1 < KQ) ? (q + 1) : (KQ - 1);       // clamped, branch-free
        v2f a1 = *(const v2f*)(Arow + qn * 4);
        v2f b1[NT];
#pragma unroll
        for (int t = 0; t < NT; ++t)
            b1[t] = *(const v2f*)(Bb + qn * (NPc * 4) + t * 64);
#pragma unroll
        for (int t = 0; t < NT; ++t)
            acc[t] = __builtin_amdgcn_wmma_f32_16x16x4_f32(
                false, a0, false, b0[t], (short)0, acc[t], false, false);
        a0 = a1;
#pragma unroll
        for (int t = 0; t < NT; ++t) b0[t] = b1[t];
    }

    const int rhalf = (lane >> 4) * 8;           // C/D: VGPR v -> M = v (+8 for hi lanes)
#pragma unroll
    for (int t = 0; t < NT; ++t) {
        const int col = n0 + t * 16 + mr;
        if ((NSTORE & 15) != 0 && col >= NSTORE) continue;   // compile-time no-op for 128
        const float badd = bias ? bias[col] : 0.0f;
#pragma unroll
        for (int v = 0; v < 8; ++v) {
            const int row = m0 + rhalf + v;
            C[(long long)row * ldc + col] = acc[t][v] + badd;
        }
    }
}

// agg[i, j] = bias[j]   (pre-folds the +b of each GCN layer before the scatter)
__global__ void init_bias_kernel(float* __restrict__ out,
                                 const float* __restrict__ bias,
                                 int width, long long total)
{
    long long i = (long long)blockIdx.x * blockDim.x + threadIdx.x;
    if (i < total) out[i] = bias[(int)(i % width)];
}

// COO SpMM scatter: agg[dst[e], :] += vals[e] * support[src[e], :]
// support fits in the 192 MB L2 (25.6 MB), so random gathers hit L2; the
// scatters are L2-side global_atomic_add_f32 (relaxed, agent scope).
template <int W>
__global__ __launch_bounds__(256) void spmm_atomic(
    const int*   __restrict__ src,
    const int*   __restrict__ dst,
    const float* __restrict__ vals,
    const float* __restrict__ sup,
    float*       __restrict__ agg)
{
    const long long total = (long long)NE * W;
    long long idx = (long long)blockIdx.x * blockDim.x + threadIdx.x;
    if (idx >= total) return;
    const int e = (int)(idx / W);
    const int j = (int)(idx - (long long)e * W);
    const float v = vals[e];
    const float x = v * sup[(long long)src[e] * W + j];
    float* p = &agg[(long long)dst[e] * W + j];
    (void)__hip_atomic_fetch_add(p, x, __ATOMIC_RELAXED, __HIP_MEMORY_SCOPE_AGENT);
}

// out[row, j] = relu(agg[row, j]) + res[row, j]; out written into its xc slice
__global__ void relu_res_kernel(const float* __restrict__ agg,
                                const float* __restrict__ res, int ldres,
                                float* __restrict__ out, int ldout)
{
    long long i = (long long)blockIdx.x * blockDim.x + threadIdx.x;
    if (i >= (long long)NN * NHID) return;
    const int row = (int)(i >> 7);               // NHID == 128
    const int j   = (int)(i & 127);
    float a = agg[i];
    a = a > 0.0f ? a : 0.0f;
    out[(long long)row * ldout + j] = a + res[(long long)row * ldres + j];
}

// Row-wise log_softmax over 40 classes: one wave32 per row, shuffle reductions.
__global__ __launch_bounds__(256) void log_softmax40(
    const float* __restrict__ in, float* __restrict__ out)
{
    const int lane = threadIdx.x & 31;
    const int wave = threadIdx.x >> 5;
    const int row  = blockIdx.x * 8 + wave;
    if (row >= NN) return;
    const float* r = in + (long long)row * NCLASS;
    const float x0 = r[lane];                                   // lane < 32 <= 40
    const float x1 = (lane + 32 < NCLASS) ? r[lane + 32] : -INFINITY;
    float m = fmaxf(x0, x1);
    for (int off = 16; off > 0; off >>= 1) m = fmaxf(m, __shfl_xor(m, off, 32));
    float s = __expf(x0 - m) + ((lane + 32 < NCLASS) ? __expf(x1 - m) : 0.0f);
    for (int off = 16; off > 0; off >>= 1) s += __shfl_xor(s, off, 32);
    const float lse = m + __logf(s);
    out[(long long)row * NCLASS + lane] = x0 - lse;
    if (lane + 32 < NCLASS) out[(long long)row * NCLASS + lane + 32] = x1 - lse;
}

// ---------------------------------------------------------------------------
extern "C" void kernel_launch(void* const* d_in, const int* in_sizes, int n_in,
                              void* d_out, int out_size, void* d_ws, size_t ws_size,
                              hipStream_t stream)
{
    const float* x      = (const float*)d_in[0];   // [50000,500]
    const float* vals   = (const float*)d_in[1];   // [1.6M]
    const float* W1     = (const float*)d_in[2];   // [500,128]
    const float* b1     = (const float*)d_in[3];   // [128]
    const float* Wh     = (const float*)d_in[4];   // [8,128,128]
    const float* bh     = (const float*)d_in[5];   // [8,128]
    const float* W10    = (const float*)d_in[6];   // [1152,40]
    const float* b10    = (const float*)d_in[7];   // [40]
    const float* weight = (const float*)d_in[8];   // [500,128]
    const float* bias   = (const float*)d_in[9];   // [128]
    const int*   src    = (const int*)d_in[10];    // [1.6M]
    const int*   dst    = (const int*)d_in[11];    // [1.6M]
    float*       outp   = (float*)d_out;           // [50000,40]

    // Workspace carve-up (floats): z | support | agg | xc | sup40 | agg40 | Bp
    float* z     = (float*)d_ws;
    float* sup   = z     + (long long)NN * NHID;
    float* agg   = sup   + (long long)NN * NHID;
    float* xc    = agg   + (long long)NN * NHID;
    float* sup40 = xc    + (long long)NN * NCAT;
    float* agg40 = sup40 + (long long)NN * NCLASS;
    float* Bp    = agg40 + (long long)NN * NCLASS;   // max(500*128, 1152*48) = 64000 floats

    const dim3 blk(256);
    const dim3 gGemm128((MTILES + 7) / 8, 2);      // NT=4 -> 64 cols per wave, 2 col groups
    const dim3 gGemm40 ((MTILES + 7) / 8, 1);      // NT=3 -> 48 cols (stores guarded to 40)

    const long long elems128 = (long long)NN * NHID;       // 6.4e6
    const long long elems40  = (long long)NN * NCLASS;     // 2.0e6
    const int bEl128 = (int)((elems128 + 255) / 256);
    const int bEl40  = (int)((elems40  + 255) / 256);
    const int bSp128 = (int)(((long long)NE * NHID   + 255) / 256);
    const int bSp40  = (int)(((long long)NE * NCLASS + 255) / 256);
    const int bRp500 = (500 * 128 + 255) / 256;
    const int bRp128 = (128 * 128 + 255) / 256;
    const int bRp40  = (1152 * NPAD40 + 255) / 256;

    // z = x @ weight + bias
    repack_B_kernel<<<bRp500, blk, 0, stream>>>(weight, NFEAT, NHID, NHID, Bp);
    gemm_wmma_f32<4, NFEAT, NHID, NHID><<<gGemm128, blk, 0, stream>>>(x, NFEAT, Bp, z, NHID, bias);

    // ---- layer 1: h1 = relu(spmm(x@W1) + b1) + z  -> xc cols [1024,1152)
    repack_B_kernel<<<bRp500, blk, 0, stream>>>(W1, NFEAT, NHID, NHID, Bp);
    gemm_wmma_f32<4, NFEAT, NHID, NHID><<<gGemm128, blk, 0, stream>>>(x, NFEAT, Bp, sup, NHID, nullptr);
    init_bias_kernel<<<bEl128, blk, 0, stream>>>(agg, b1, NHID, elems128);
    spmm_atomic<NHID><<<bSp128, blk, 0, stream>>>(src, dst, vals, sup, agg);
    relu_res_kernel<<<bEl128, blk, 0, stream>>>(agg, z, NHID, xc + 8 * NHID, NCAT);

    // ---- layers 2..9: h_{k+1} = relu(spmm(h_k@Wh[i]) + bh[i]) + h_k
    // x_k lives at xc columns (9-k)*128
    for (int i = 0; i < 8; ++i) {
        const float* hin  = xc + (8 - i) * NHID;   // x_{i+1}
        float*       hout = xc + (7 - i) * NHID;   // x_{i+2}
        repack_B_kernel<<<bRp128, blk, 0, stream>>>(Wh + (long long)i * NHID * NHID,
                                                    NHID, NHID, NHID, Bp);
        gemm_wmma_f32<4, NHID, NHID, NHID><<<gGemm128, blk, 0, stream>>>(hin, NCAT, Bp, sup, NHID, nullptr);
        init_bias_kernel<<<bEl128, blk, 0, stream>>>(agg, bh + i * NHID, NHID, elems128);
        spmm_atomic<NHID><<<bSp128, blk, 0, stream>>>(src, dst, vals, sup, agg);
        relu_res_kernel<<<bEl128, blk, 0, stream>>>(agg, hin, NCAT, hout, NCAT);
    }

    // ---- classifier: out = log_softmax(spmm(xc@W10) + b10)
    repack_B_kernel<<<bRp40, blk, 0, stream>>>(W10, NCAT, NCLASS, NPAD40, Bp);
    gemm_wmma_f32<3, NCAT, NCLASS, NPAD40><<<gGemm40, blk, 0, stream>>>(xc, NCAT, Bp, sup40, NCLASS, nullptr);
    init_bias_kernel<<<bEl40, blk, 0, stream>>>(agg40, b10, NCLASS, elems40);
    spmm_atomic<NCLASS><<<bSp40, blk, 0, stream>>>(src, dst, vals, sup40, agg40);
    log_softmax40<<<(NN + 7) / 8, blk, 0, stream>>>(agg40, outp);
}